// Network_47699906789920
// MI455X (gfx1250) — compile-verified
//
#include <hip/hip_runtime.h>
#include <math.h>

#define HS   4096
#define NIN  17
#define NACT 4

typedef __attribute__((ext_vector_type(2))) float v2f;
typedef __attribute__((ext_vector_type(4))) float v4f;
typedef __attribute__((ext_vector_type(8))) float v8f;

// ---------------------------------------------------------------------------
// K0: pre[j] = i2h_b[j] + sum_n inputs[n] * i2h_w[j*NIN + n]
// ---------------------------------------------------------------------------
__global__ void k_init_pre(const float* __restrict__ inputs,
                           const float* __restrict__ i2h_w,
                           const float* __restrict__ i2h_b,
                           float* __restrict__ pre) {
    int j = blockIdx.x * blockDim.x + threadIdx.x;
    if (j >= HS) return;
    float s = i2h_b[j];
    const float* wr = i2h_w + (size_t)j * NIN;
#pragma unroll
    for (int n = 0; n < NIN; ++n) s = fmaf(inputs[n], wr[n], s);
    pre[j] = s;
}

// ---------------------------------------------------------------------------
// K1: pre[j] += sum_i hidden[i] * (w[i,j] + alpha[i,j]*hebb[i,j])
// via V_WMMA_F32_16X16X4_F32. A = hidden chunk replicated into all 16 rows,
// B = on-the-fly (w + alpha*hebb) 4x16 tile. All D rows identical partials;
// row 0 (lanes 0..15, c[0]) is atomically accumulated into pre[].
// w/alpha are dead after this kernel -> non-temporal loads, so hebb (64 MB)
// can stay resident in the 192 MB L2 for K4's re-read.
// ---------------------------------------------------------------------------
#define KSPLIT          32
#define ROWS_PER_CHUNK  (HS / KSPLIT)   // 128 rows per wave -> 32 WMMA iters
#define WAVES_PER_BLOCK 4
#define COL_TILES       (HS / 16)       // 256

__global__ void k_matvec_wmma(const float* __restrict__ hidden,
                              const float* __restrict__ w,
                              const float* __restrict__ alpha,
                              const float* __restrict__ hebb,
                              float* __restrict__ pre) {
    const int lane   = threadIdx.x & 31;
    const int gwave  = blockIdx.x * WAVES_PER_BLOCK + (threadIdx.x >> 5);
    const int colTile = gwave & (COL_TILES - 1);     // 0..255
    const int kchunk  = gwave >> 8;                  // 0..KSPLIT-1
    const int j0   = colTile << 4;
    const int kbeg = kchunk * ROWS_PER_CHUNK;
    const int half = lane >> 4;                      // K-pair selector (0: K0/1, 1: K2/3)
    const int col  = lane & 15;

    v8f c = {};
    const size_t ld = HS;
    for (int k = kbeg; k < kbeg + ROWS_PER_CHUNK; k += 4) {
        const int r = k + 2 * half;
        const size_t off0 = (size_t)r * ld + (size_t)(j0 + col);
        const size_t off1 = off0 + ld;
        // w/alpha: non-temporal (single use); hebb: regular (keep in L2 for K4)
        const float w0 = __builtin_nontemporal_load(w + off0);
        const float w1 = __builtin_nontemporal_load(w + off1);
        const float a0 = __builtin_nontemporal_load(alpha + off0);
        const float a1 = __builtin_nontemporal_load(alpha + off1);
        v2f b;
        b.x = fmaf(a0, hebb[off0], w0);              // row r   of (w + a*hebb)
        b.y = fmaf(a1, hebb[off1], w1);              // row r+1
        v2f a;
        a.x = hidden[r];                             // replicated across all M rows
        a.y = hidden[r + 1];
        // D = A(16x4) * B(4x16) + C   (f32 native WMMA)
        c = __builtin_amdgcn_wmma_f32_16x16x4_f32(
                /*neg_a=*/false, a, /*neg_b=*/false, b,
                /*c_mod=*/(short)0, c, /*reuse_a=*/false, /*reuse_b=*/false);
    }
    // D row 0: VGPR0, lanes 0..15 hold pre-partials for columns j0..j0+15.
    if (lane < 16) atomicAdd(&pre[j0 + lane], c[0]);
}

// ---------------------------------------------------------------------------
// K2: hactiv[j] = tanh(pre[j]); written to d_out (4B-aligned region) AND to
// an aligned workspace copy so K3/K4 can use vector loads.
// ---------------------------------------------------------------------------
__global__ void k_tanh_act(const float* __restrict__ pre,
                           float* __restrict__ hactiv_out,
                           float* __restrict__ hactiv_ws) {
    int j = blockIdx.x * blockDim.x + threadIdx.x;
    if (j >= HS) return;
    const float h = tanhf(pre[j]);
    hactiv_out[j] = h;
    hactiv_ws[j]  = h;
}

// ---------------------------------------------------------------------------
// K3: output heads. Block b<4: activout[b] = hactiv . h2o_w[b,:] + h2o_b[b]
//     Block 4:   valueout    = hactiv . h2v_w      + h2v_b
// ---------------------------------------------------------------------------
__global__ void k_heads(const float* __restrict__ hactiv,
                        const float* __restrict__ h2o_w,
                        const float* __restrict__ h2o_b,
                        const float* __restrict__ h2v_w,
                        const float* __restrict__ h2v_b,
                        float* __restrict__ out) {
    __shared__ float red[256];
    const int b = blockIdx.x;                       // 0..4
    const float* wrow = (b < NACT) ? (h2o_w + (size_t)b * HS) : h2v_w;
    float s = 0.0f;
    for (int j = threadIdx.x; j < HS; j += blockDim.x)
        s = fmaf(hactiv[j], wrow[j], s);
    red[threadIdx.x] = s;
    __syncthreads();
    for (int off = 128; off > 0; off >>= 1) {
        if ((int)threadIdx.x < off) red[threadIdx.x] += red[threadIdx.x + off];
        __syncthreads();
    }
    if (threadIdx.x == 0)
        out[b] = red[0] + ((b < NACT) ? h2o_b[b] : h2v_b[0]);
}

// ---------------------------------------------------------------------------
// K4: hebb_new[i,j] = (1-eta)*hebb[i,j] + eta*hidden[i]*hactiv[j]
// hebb read as aligned B128 (should hit L2 after K1); stores are scalar
// non-temporal (dest region is only 4B-aligned, and the data is write-once).
// ---------------------------------------------------------------------------
__global__ void k_hebb_update(const float* __restrict__ hebb,
                              const float* __restrict__ hidden,
                              const float* __restrict__ hactiv,   // aligned ws copy
                              const float* __restrict__ eta_p,
                              float* __restrict__ out_hebb) {
    const float eta = eta_p[0];
    const float om  = 1.0f - eta;
    const size_t nquads = (size_t)HS * (size_t)HS / 4;   // 4,194,304
    const size_t stride = (size_t)gridDim.x * blockDim.x;
    for (size_t q = (size_t)blockIdx.x * blockDim.x + threadIdx.x;
         q < nquads; q += stride) {
        const size_t idx = q * 4;
        const int i = (int)(idx >> 12);          // row  (hidden index)
        const int j = (int)(idx & (HS - 1));     // col  (hactiv index), %4 == 0
        const v4f h  = *(const v4f*)(hebb + idx);        // aligned B128
        const v4f av = *(const v4f*)(hactiv + j);        // aligned B128 (ws copy)
        const float hi = hidden[i] * eta;
        __builtin_nontemporal_store(fmaf(om, h.x, hi * av.x), out_hebb + idx + 0);
        __builtin_nontemporal_store(fmaf(om, h.y, hi * av.y), out_hebb + idx + 1);
        __builtin_nontemporal_store(fmaf(om, h.z, hi * av.z), out_hebb + idx + 2);
        __builtin_nontemporal_store(fmaf(om, h.w, hi * av.w), out_hebb + idx + 3);
    }
}

// ---------------------------------------------------------------------------
extern "C" void kernel_launch(void* const* d_in, const int* in_sizes, int n_in,
                              void* d_out, int out_size, void* d_ws, size_t ws_size,
                              hipStream_t stream) {
    const float* inputs = (const float*)d_in[0];
    const float* hidden = (const float*)d_in[1];
    const float* hebb   = (const float*)d_in[2];
    const float* et     = (const float*)d_in[3];
    const float* pw     = (const float*)d_in[4];
    const float* i2h_w  = (const float*)d_in[5];
    const float* i2h_b  = (const float*)d_in[6];
    const float* w      = (const float*)d_in[7];
    const float* alpha  = (const float*)d_in[8];
    const float* eta    = (const float*)d_in[9];
    const float* h2o_w  = (const float*)d_in[10];
    const float* h2o_b  = (const float*)d_in[11];
    const float* h2v_w  = (const float*)d_in[12];
    const float* h2v_b  = (const float*)d_in[13];

    float* out         = (float*)d_out;
    float* out_heads   = out;                       // activout[4] then valueout[1]
    float* out_hactiv  = out + 5;                   // 4096
    float* out_hebb    = out + 5 + HS;              // 4096*4096 (4B-aligned only)
    float* out_et      = out_hebb + (size_t)HS * HS;
    float* out_pw      = out_et   + (size_t)HS * HS;

    float* pre       = (float*)d_ws;                // 4096 f32 scratch
    float* hactiv_ws = pre + HS;                    // 4096 f32, 16B-aligned

    // 1. pre = i2h(inputs) + bias
    k_init_pre<<<HS / 256, 256, 0, stream>>>(inputs, i2h_w, i2h_b, pre);

    // 2. pre += hidden @ (w + alpha*hebb)   [WMMA f32 16x16x4, k-split x32]
    const int nblocks = (COL_TILES * KSPLIT) / WAVES_PER_BLOCK;  // 2048
    k_matvec_wmma<<<nblocks, WAVES_PER_BLOCK * 32, 0, stream>>>(
        hidden, w, alpha, hebb, pre);

    // 3. hactiv = tanh(pre)
    k_tanh_act<<<HS / 256, 256, 0, stream>>>(pre, out_hactiv, hactiv_ws);

    // 4. heads
    k_heads<<<NACT + 1, 256, 0, stream>>>(hactiv_ws, h2o_w, h2o_b, h2v_w, h2v_b,
                                          out_heads);

    // 5. hebb update (streaming; hebb hopefully L2-resident from step 2)
    k_hebb_update<<<4096, 256, 0, stream>>>(hebb, hidden, hactiv_ws, eta, out_hebb);

    // 6. et / pw pass-through: driver-optimized D2D copies (graph-capture safe)
    hipMemcpyAsync(out_et, et, (size_t)HS * HS * sizeof(float),
                   hipMemcpyDeviceToDevice, stream);
    hipMemcpyAsync(out_pw, pw, (size_t)HS * HS * sizeof(float),
                   hipMemcpyDeviceToDevice, stream);
}